// DegreePrediction_28905129902567
// MI455X (gfx1250) — compile-verified
//
#include <hip/hip_runtime.h>

// N = 96; streams 96^4 floats (340 MB) once -> pure HBM-bandwidth problem.
// Stage 1: 1152 blocks x 256 threads, each block reduces 8 "p" slabs of
//          weights_r into 96 per-block partial sums (LDS ds_add_f32).
// Stage 2: one block, 6 wave32s, reduces the 1152x96 partial matrix with
//          V_WMMA_F32_16X16X4_F32 (A = ones), plus the scalar mask correction.

#define NN 96
#define N2 (NN * NN)              // 9216 values of p
#define PER_BLOCK_P 8
#define NBLK (N2 / PER_BLOCK_P)   // 1152 blocks
#define THREADS 256
#define F4_PER_P (N2 / 4)         // 2304 float4 per p-slab
#define ITERS (F4_PER_P / THREADS) // 9

typedef __attribute__((ext_vector_type(4))) float f4;
typedef __attribute__((ext_vector_type(2))) float v2f;
typedef __attribute__((ext_vector_type(8))) float v8f;

__global__ __launch_bounds__(THREADS)
void dp_partial(const float* __restrict__ x, const float* __restrict__ wt,
                const float* __restrict__ R, float* __restrict__ ws) {
    __shared__ float l2s[PER_BLOCK_P];
    __shared__ float ypart[NN];
    const int t = threadIdx.x;
    const int pbase = blockIdx.x * PER_BLOCK_P;

    if (t < PER_BLOCK_P) {
        const int p = pbase + t;
        float v = x[p] * wt[p];
        if (p == 0 || p == NN + 1) v = 0.0f;   // masked diag entries -> l2 = 0
        l2s[t] = v;
    }
    if (t < NN) ypart[t] = 0.0f;
    __syncthreads();

    float acc[ITERS];
#pragma unroll
    for (int m = 0; m < ITERS; ++m) acc[m] = 0.0f;

    const f4* Rb = ((const f4*)R) + (size_t)pbase * F4_PER_P;
    for (int pp = 0; pp < PER_BLOCK_P; ++pp) {
        const f4* row = Rb + (size_t)pp * F4_PER_P;
        // hint the next p-slab into cache (emits global_prefetch_b8)
        __builtin_prefetch(row + F4_PER_P + t, 0, 0);
        const float l2v = l2s[pp];
#pragma unroll
        for (int m = 0; m < ITERS; ++m) {
            // single-use stream: non-temporal b128 load, don't pollute L2
            f4 r = __builtin_nontemporal_load(&row[t + THREADS * m]);
            acc[m] = fmaf(l2v, (r.x + r.y) + (r.z + r.w), acc[m]);
        }
    }

    // q = t + 256*m lies in a fixed (k = q/24) row for every p-slab
#pragma unroll
    for (int m = 0; m < ITERS; ++m) {
        const int k = (t + THREADS * m) / (NN / 4);
        atomicAdd(&ypart[k], acc[m]);           // ds_add_f32
    }
    __syncthreads();
    if (t < NN) ws[(size_t)blockIdx.x * NN + t] = ypart[t];
}

// Reduce ws (NBLK x 96) column-wise with WMMA: D = ones(16x4) * B(4x16) + C,
// accumulated over NBLK/4 chunks; row M=0 of D holds the column sums.
__global__ __launch_bounds__(192)
void dp_reduce(const float* __restrict__ x, const float* __restrict__ wt,
               const float* __restrict__ R, const float* __restrict__ ws,
               float* __restrict__ out) {
    const int lane = threadIdx.x & 31;
    const int wave = threadIdx.x >> 5;          // 0..5 -> k tile
    const int k0 = wave * 16;
    const int rsel = (lane < 16) ? 0 : 2;       // B 4x16 layout: v0 = rows {0,2}, v1 = rows {1,3}
    const int col = k0 + (lane & 15);

    v2f a; a.x = 1.0f; a.y = 1.0f;              // A = ones(16x4)
    v8f acc = {};
    for (int b = 0; b < NBLK; b += 4) {
        const float* bp = ws + (size_t)(b + rsel) * NN + col;
        v2f bv; bv.x = bp[0]; bv.y = bp[NN];
        acc = __builtin_amdgcn_wmma_f32_16x16x4_f32(
            /*neg_a=*/false, a, /*neg_b=*/false, bv,
            /*c_mod=*/(short)0, acc, /*reuse_a=*/false, /*reuse_b=*/false);
    }

    if (lane < 16) {
        const int k = k0 + lane;
        float y = acc[0];                       // C/D layout: VGPR0, lanes 0-15 = (M=0, N=lane)
        if (k < 2) {
            // Correction for the 16 overwritten entries of weights_r.
            // Only (i,j) = (0,1) [p=1] and (1,0) [p=96] matter (l2=0 elsewhere).
            const float l2_01 = x[1] * wt[1];
            const float l2_10 = x[NN] * wt[NN];
#pragma unroll
            for (int dd = 0; dd < 2; ++dd) {
                const float fix01 = (k == 0 && dd == 0) ? 1.0f : 0.0f; // (0,1,0,0) -> 1
                const float fix10 = (k == 1 && dd == 1) ? 1.0f : 0.0f; // (1,0,1,1) -> 1
                y += l2_01 * (fix01 - R[((size_t)(1 * NN + k)) * NN + dd]);
                y += l2_10 * (fix10 - R[((size_t)(NN * NN + k)) * NN + dd]);
            }
        }
        out[k] = y;
    }
}

extern "C" void kernel_launch(void* const* d_in, const int* in_sizes, int n_in,
                              void* d_out, int out_size, void* d_ws, size_t ws_size,
                              hipStream_t stream) {
    const float* x  = (const float*)d_in[0];
    // d_in[1] = weights_t_fix: mathematically unused (its selected entries are zeroed)
    const float* wt = (const float*)d_in[2];
    const float* R  = (const float*)d_in[3];
    float* out = (float*)d_out;
    float* ws  = (float*)d_ws;   // needs NBLK*96*4 = 442368 bytes

    dp_partial<<<NBLK, THREADS, 0, stream>>>(x, wt, R, ws);
    dp_reduce<<<1, 192, 0, stream>>>(x, wt, R, ws, out);
}